// Net_30485677867755
// MI455X (gfx1250) — compile-verified
//
#include <hip/hip_runtime.h>
#include <hip/hip_bf16.h>
#include <math.h>

// GCN 2-layer forward for MI455X (gfx1250, wave32).
// Pipeline: deg -> dinv -> edge weights -> GEMM1 (WMMA f32 16x16x4) ->
// scatter-agg1 -> self+bias+ELU+BN-stats -> BN finalize ->
// GEMM2 (WMMA, BN+ReLU fused on A) -> scatter-agg2 -> self+bias -> log_softmax.
// LDS weight layout is K-pair interleaved (float2) so each WMMA B fragment is a
// single aligned ds_load_b64 (no VALU repacking in the inner loop).

typedef float v2f __attribute__((ext_vector_type(2)));
typedef float v8f __attribute__((ext_vector_type(8)));

#define IN_CH 256
#define HID_CH 32
#define OUT_CH 40
#define OUT_PAD 48
#define BN_EPS 1e-5f

// ---------------- degree / normalization ----------------

__global__ __launch_bounds__(256) void k_init_deg(float* deg, int n) {
    int i = blockIdx.x * 256 + threadIdx.x;
    if (i < n) deg[i] = 1.0f;            // self-loop contributes 1
}

__global__ __launch_bounds__(256) void k_deg_accum(float* deg, const int* dst, int e) {
    int i = blockIdx.x * 256 + threadIdx.x;
    if (i < e) atomicAdd(&deg[dst[i]], 1.0f);
}

__global__ __launch_bounds__(256) void k_dinv(float* deg, int n) {
    int i = blockIdx.x * 256 + threadIdx.x;
    if (i < n) deg[i] = rsqrtf(deg[i]);  // deg >= 1 always
}

__global__ __launch_bounds__(256) void k_edge_w(float* ewt, const float* dinv,
                                                const int* src, const int* dst, int e) {
    int i = blockIdx.x * 256 + threadIdx.x;
    if (i < e) ewt[i] = dinv[src[i]] * dinv[dst[i]];
}

__global__ __launch_bounds__(256) void k_build_w2pad(float* w2pad, const float* W2) {
    int i = blockIdx.x * 256 + threadIdx.x;           // 32*48
    if (i < HID_CH * OUT_PAD) {
        int k = i / OUT_PAD, c = i - k * OUT_PAD;
        w2pad[i] = (c < OUT_CH) ? W2[k * OUT_CH + c] : 0.0f;
    }
}

// ---------------- GEMM1: h1 = x @ W1  [N,256]x[256,32] ----------------
// 256 threads = 8 waves/block; each wave owns 16 rows, computes both 16-col tiles.
// LDS: w1v[p*32 + c] = {W1[2p][c], W1[2p+1][c]}  (p = K-pair 0..127, c = 0..31)

__global__ __launch_bounds__(256) void k_gemm1_wmma(const float* __restrict__ x,
                                                    const float* __restrict__ W1,
                                                    float* __restrict__ h1, int n) {
    __shared__ float2 w1v[(IN_CH / 2) * HID_CH];       // 32 KB
    for (int i = threadIdx.x; i < (IN_CH / 2) * HID_CH; i += 256) {
        int p = i >> 5, c = i & 31;                    // K-pair, column
        float2 t;
        t.x = W1[(2 * p) * HID_CH + c];
        t.y = W1[(2 * p + 1) * HID_CH + c];
        w1v[i] = t;
    }
    __syncthreads();

    const int wave = threadIdx.x >> 5, lane = threadIdx.x & 31;
    const int row0 = (blockIdx.x * 8 + wave) * 16;
    if (row0 >= n) return;                             // uniform per wave -> EXEC stays full

    const int half = lane >> 4;                        // 0: K={k,k+1}, 1: K={k+2,k+3}
    const int r = lane & 15;                           // A row / B,D column
    const int arow = (row0 + r < n) ? (row0 + r) : (n - 1);  // tail clamp (loads only)
    const float* xp = x + (size_t)arow * IN_CH + 2 * half;

    v8f acc0 = {0.f,0.f,0.f,0.f,0.f,0.f,0.f,0.f};
    v8f acc1 = {0.f,0.f,0.f,0.f,0.f,0.f,0.f,0.f};

    for (int k = 0; k < IN_CH; k += 4) {
        v2f a; a.x = xp[k]; a.y = xp[k + 1];
        const int p = (k >> 1) + half;                 // K-pair index for this lane half
        float2 f0 = w1v[p * HID_CH + r];               // single ds_load_b64
        float2 f1 = w1v[p * HID_CH + r + 16];
        v2f b0; b0.x = f0.x; b0.y = f0.y;
        v2f b1; b1.x = f1.x; b1.y = f1.y;
        acc0 = __builtin_amdgcn_wmma_f32_16x16x4_f32(false, a, false, b0, (short)0, acc0, false, false);
        acc1 = __builtin_amdgcn_wmma_f32_16x16x4_f32(false, a, false, b1, (short)0, acc1, false, false);
    }
    for (int j = 0; j < 8; ++j) {
        int m = row0 + j + 8 * half;
        if (m < n) {
            h1[(size_t)m * HID_CH + r]      = acc0[j];
            h1[(size_t)m * HID_CH + r + 16] = acc1[j];
        }
    }
}

// ---------------- layer-1 aggregation ----------------

__global__ __launch_bounds__(256) void k_scatter1(const float* __restrict__ h1,
                                                  float* __restrict__ agg1,
                                                  const float* __restrict__ ewt,
                                                  const int* __restrict__ src,
                                                  const int* __restrict__ dst,
                                                  long long total) {
    long long tid = (long long)blockIdx.x * 256 + threadIdx.x;
    if (tid >= total) return;
    int e = (int)(tid >> 5), c = (int)(tid & 31);
    float w = ewt[e];
    atomicAdd(&agg1[(size_t)dst[e] * HID_CH + c], w * h1[(size_t)src[e] * HID_CH + c]);
}

// self-loop + bias + ELU (in place into agg1) + per-block BN stat reduction
__global__ __launch_bounds__(256) void k_self1_elu_bn(const float* __restrict__ h1,
                                                      float* __restrict__ agg1,
                                                      const float* __restrict__ dinv,
                                                      const float* __restrict__ b1,
                                                      float* __restrict__ bnsum,
                                                      float* __restrict__ bnsumsq,
                                                      long long total) {
    __shared__ float sv[256], sv2[256];
    long long tid = (long long)blockIdx.x * 256 + threadIdx.x;
    float v = 0.0f;
    if (tid < total) {
        int i = (int)(tid >> 5), c = (int)(tid & 31);
        float di = dinv[i];
        v = agg1[tid] + di * di * h1[tid] + b1[c];
        v = (v > 0.0f) ? v : expm1f(v);                // ELU(alpha=1)
        agg1[tid] = v;
    }
    sv[threadIdx.x] = v; sv2[threadIdx.x] = v * v;
    __syncthreads();
    if (threadIdx.x < 32) {
        float s = 0.f, s2 = 0.f;
        for (int j = 0; j < 8; ++j) { s += sv[threadIdx.x + 32 * j]; s2 += sv2[threadIdx.x + 32 * j]; }
        atomicAdd(&bnsum[threadIdx.x], s);
        atomicAdd(&bnsumsq[threadIdx.x], s2);
    }
}

__global__ void k_bn_finalize(const float* bnsum, const float* bnsumsq,
                              float* bnmean, float* bnscale, int n) {
    int c = threadIdx.x;
    if (c < HID_CH) {
        float mean = bnsum[c] / (float)n;
        float var = bnsumsq[c] / (float)n - mean * mean;  // biased variance
        bnmean[c] = mean;
        bnscale[c] = rsqrtf(var + BN_EPS);
    }
}

// ---------------- GEMM2: pre2 = relu(bn(h1e)) @ W2  [N,32]x[32,40->48] ----------------
// LDS: w2v[p*48 + c] = {W2p[2p][c], W2p[2p+1][c]}  (p = 0..15, c = 0..47)

__global__ __launch_bounds__(256) void k_gemm2_wmma(const float* __restrict__ h1e,
                                                    const float* __restrict__ w2pad,
                                                    const float* __restrict__ bnmean,
                                                    const float* __restrict__ bnscale,
                                                    float* __restrict__ pre2, int n) {
    __shared__ float2 w2v[(HID_CH / 2) * OUT_PAD];     // 6 KB
    __shared__ float m_s[HID_CH], s_s[HID_CH];
    for (int i = threadIdx.x; i < (HID_CH / 2) * OUT_PAD; i += 256) {
        int p = i / OUT_PAD, c = i - p * OUT_PAD;
        float2 t;
        t.x = w2pad[(2 * p) * OUT_PAD + c];
        t.y = w2pad[(2 * p + 1) * OUT_PAD + c];
        w2v[i] = t;
    }
    if (threadIdx.x < HID_CH) {
        m_s[threadIdx.x] = bnmean[threadIdx.x];
        s_s[threadIdx.x] = bnscale[threadIdx.x];
    }
    __syncthreads();

    const int wave = threadIdx.x >> 5, lane = threadIdx.x & 31;
    const int row0 = (blockIdx.x * 8 + wave) * 16;
    if (row0 >= n) return;

    const int half = lane >> 4;
    const int r = lane & 15;
    const int arow = (row0 + r < n) ? (row0 + r) : (n - 1);
    const float* ap = h1e + (size_t)arow * HID_CH + 2 * half;

    v8f zero = {0.f,0.f,0.f,0.f,0.f,0.f,0.f,0.f};
    v8f acc[3]; acc[0] = zero; acc[1] = zero; acc[2] = zero;

    for (int k = 0; k < HID_CH; k += 4) {
        const int kk = k + 2 * half;
        float a0 = (ap[k]     - m_s[kk])     * s_s[kk];
        float a1 = (ap[k + 1] - m_s[kk + 1]) * s_s[kk + 1];
        v2f a; a.x = (a0 > 0.f) ? a0 : 0.f; a.y = (a1 > 0.f) ? a1 : 0.f;  // fused BN+ReLU
        const int p = (k >> 1) + half;
        for (int t = 0; t < 3; ++t) {
            float2 f = w2v[p * OUT_PAD + r + 16 * t];  // single ds_load_b64
            v2f b; b.x = f.x; b.y = f.y;
            acc[t] = __builtin_amdgcn_wmma_f32_16x16x4_f32(false, a, false, b, (short)0, acc[t], false, false);
        }
    }
    for (int j = 0; j < 8; ++j) {
        int m = row0 + j + 8 * half;
        if (m < n) {
            for (int t = 0; t < 3; ++t) {
                int c = r + 16 * t;
                if (c < OUT_CH) pre2[(size_t)m * OUT_CH + c] = acc[t][j];
            }
        }
    }
}

// ---------------- layer-2 aggregation + log_softmax ----------------

__global__ __launch_bounds__(256) void k_scatter2(const float* __restrict__ pre2,
                                                  float* __restrict__ out,
                                                  const float* __restrict__ ewt,
                                                  const int* __restrict__ src,
                                                  const int* __restrict__ dst,
                                                  long long total) {
    long long tid = (long long)blockIdx.x * 256 + threadIdx.x;
    if (tid >= total) return;
    int e = (int)(tid / OUT_CH), c = (int)(tid - (long long)e * OUT_CH);
    float w = ewt[e];
    atomicAdd(&out[(size_t)dst[e] * OUT_CH + c], w * pre2[(size_t)src[e] * OUT_CH + c]);
}

__global__ __launch_bounds__(256) void k_self2(const float* __restrict__ pre2,
                                               float* __restrict__ out,
                                               const float* __restrict__ dinv,
                                               const float* __restrict__ b2,
                                               long long total) {
    long long tid = (long long)blockIdx.x * 256 + threadIdx.x;
    if (tid >= total) return;
    int i = (int)(tid / OUT_CH), c = (int)(tid - (long long)i * OUT_CH);
    float di = dinv[i];
    out[tid] += di * di * pre2[tid] + b2[c];
}

// one wave32 per node; lane handles channels {lane, lane+32(<40)}
__global__ __launch_bounds__(256) void k_logsoftmax(float* __restrict__ out, int n) {
    int wave = threadIdx.x >> 5, lane = threadIdx.x & 31;
    int i = blockIdx.x * 8 + wave;
    if (i >= n) return;
    float* row = out + (size_t)i * OUT_CH;
    float v0 = row[lane];
    float v1 = (lane < 8) ? row[32 + lane] : -INFINITY;
    float m = fmaxf(v0, v1);
    for (int off = 16; off; off >>= 1) m = fmaxf(m, __shfl_xor(m, off, 32));
    float s = expf(v0 - m) + ((lane < 8) ? expf(v1 - m) : 0.0f);
    for (int off = 16; off; off >>= 1) s += __shfl_xor(s, off, 32);
    float lse = m + logf(s);
    row[lane] = v0 - lse;
    if (lane < 8) row[32 + lane] = v1 - lse;
}

// ---------------- launch ----------------

extern "C" void kernel_launch(void* const* d_in, const int* in_sizes, int n_in,
                              void* d_out, int out_size, void* d_ws, size_t ws_size,
                              hipStream_t stream) {
    const float* x  = (const float*)d_in[0];
    const int*   ei = (const int*)d_in[1];
    const float* W1 = (const float*)d_in[2];
    const float* b1 = (const float*)d_in[3];
    const float* W2 = (const float*)d_in[4];
    const float* b2 = (const float*)d_in[5];
    float* out = (float*)d_out;

    const int N = in_sizes[0] / IN_CH;
    const int E = in_sizes[1] / 2;
    const int* src = ei;
    const int* dst = ei + E;

    // workspace partition (floats)
    float* ws = (float*)d_ws;
    float* dinv   = ws;                         size_t off = (size_t)N;
    float* h1     = ws + off;                   off += (size_t)N * HID_CH;
    float* agg1   = ws + off;                   off += (size_t)N * HID_CH;   // becomes h1e (post-ELU)
    float* pre2   = ws + off;                   off += (size_t)N * OUT_CH;
    float* ewt    = ws + off;                   off += (size_t)E;
    float* w2pad  = ws + off;                   off += (size_t)HID_CH * OUT_PAD;
    float* bnsum  = ws + off;                   off += 32;
    float* bnsumsq= ws + off;                   off += 32;
    float* bnmean = ws + off;                   off += 32;
    float* bnscale= ws + off;                   off += 32;

    // zero the atomic accumulators (graph-capturable memset nodes)
    hipMemsetAsync(agg1, 0, (size_t)N * HID_CH * sizeof(float), stream);
    hipMemsetAsync(out,  0, (size_t)N * OUT_CH * sizeof(float), stream);
    hipMemsetAsync(bnsum, 0, 64 * sizeof(float), stream);   // bnsum + bnsumsq contiguous

    const int nb_n   = (N + 255) / 256;
    const int nb_e   = (E + 255) / 256;
    const int nb_g   = (N + 127) / 128;                     // 8 waves x 16 rows per block
    const long long t1 = (long long)E * HID_CH;
    const long long t2 = (long long)E * OUT_CH;
    const long long n1 = (long long)N * HID_CH;
    const long long n2 = (long long)N * OUT_CH;

    k_init_deg  <<<nb_n, 256, 0, stream>>>(dinv, N);
    k_deg_accum <<<nb_e, 256, 0, stream>>>(dinv, dst, E);
    k_dinv      <<<nb_n, 256, 0, stream>>>(dinv, N);
    k_edge_w    <<<nb_e, 256, 0, stream>>>(ewt, dinv, src, dst, E);
    k_build_w2pad<<<(HID_CH * OUT_PAD + 255) / 256, 256, 0, stream>>>(w2pad, W2);

    k_gemm1_wmma<<<nb_g, 256, 0, stream>>>(x, W1, h1, N);

    k_scatter1  <<<(int)((t1 + 255) / 256), 256, 0, stream>>>(h1, agg1, ewt, src, dst, t1);
    k_self1_elu_bn<<<(int)((n1 + 255) / 256), 256, 0, stream>>>(h1, agg1, dinv, b1, bnsum, bnsumsq, n1);
    k_bn_finalize<<<1, 32, 0, stream>>>(bnsum, bnsumsq, bnmean, bnscale, N);

    k_gemm2_wmma<<<nb_g, 256, 0, stream>>>(agg1, w2pad, bnmean, bnscale, pre2, N);

    k_scatter2  <<<(int)((t2 + 255) / 256), 256, 0, stream>>>(pre2, out, ewt, src, dst, t2);
    k_self2     <<<(int)((n2 + 255) / 256), 256, 0, stream>>>(pre2, out, dinv, b2, n2);
    k_logsoftmax<<<(N + 7) / 8, 256, 0, stream>>>(out, N);
}